// VybnLinguaV3_67972152426770
// MI455X (gfx1250) — compile-verified
//
#include <hip/hip_runtime.h>
#include <stdint.h>

typedef float    v2f __attribute__((ext_vector_type(2)));
typedef float    v4f __attribute__((ext_vector_type(4)));
typedef float    v8f __attribute__((ext_vector_type(8)));
typedef unsigned v4u __attribute__((ext_vector_type(4)));
typedef int      v4i __attribute__((ext_vector_type(4)));
typedef int      v8i __attribute__((ext_vector_type(8)));

#define NPRIM 1024
#define DIM   1024
#define ROWS_TOTAL (8 * 4096)          // 32768 (b,s) rows
#define ROWS_HALF  (ROWS_TOTAL / 2)    // 16384 : threefry counter pairing distance (in rows)
#define N_ELEMS_HALF 16777216u         // (ROWS_TOTAL*NPRIM)/2

// ---------------- exact JAX threefry-2x32 (key = jax.random.key(42) -> (0,42)) ----------------
__device__ __forceinline__ void tf_round(unsigned& x0, unsigned& x1, const int r) {
  x0 += x1;
  x1 = (x1 << r) | (x1 >> (32 - r));
  x1 ^= x0;
}

__device__ __forceinline__ void threefry2x32_42(unsigned i0, unsigned i1,
                                                unsigned& o0, unsigned& o1) {
  const unsigned ks0 = 0u, ks1 = 42u;
  const unsigned ks2 = 0x1BD11BDAu ^ ks0 ^ ks1;
  unsigned x0 = i0 + ks0, x1 = i1 + ks1;
  tf_round(x0, x1, 13); tf_round(x0, x1, 15); tf_round(x0, x1, 26); tf_round(x0, x1, 6);
  x0 += ks1; x1 += ks2 + 1u;
  tf_round(x0, x1, 17); tf_round(x0, x1, 29); tf_round(x0, x1, 16); tf_round(x0, x1, 24);
  x0 += ks2; x1 += ks0 + 2u;
  tf_round(x0, x1, 13); tf_round(x0, x1, 15); tf_round(x0, x1, 26); tf_round(x0, x1, 6);
  x0 += ks0; x1 += ks1 + 3u;
  tf_round(x0, x1, 17); tf_round(x0, x1, 29); tf_round(x0, x1, 16); tf_round(x0, x1, 24);
  x0 += ks1; x1 += ks2 + 4u;
  tf_round(x0, x1, 13); tf_round(x0, x1, 15); tf_round(x0, x1, 26); tf_round(x0, x1, 6);
  x0 += ks2; x1 += ks0 + 5u;
  o0 = x0; o1 = x1;
}

__device__ __forceinline__ float gumbel_from_bits(unsigned b) {
  // jax uniform: bits>>9 | 0x3f800000 -> [1,2), -1 -> [0,1), affine to [tiny,1)
  const float tiny = 1.1754943508222875e-38f;
  float f = __uint_as_float((b >> 9) | 0x3f800000u) - 1.0f;
  float u = fmaxf(tiny, f * (1.0f - tiny) + tiny);
  return -__logf(-__logf(u));
}

// ---------------- kernels ----------------
__global__ void vybn_zero_counts(int* __restrict__ counts) {
  counts[threadIdx.x] = 0;
}

__device__ __forceinline__ void vybn_issue_tdm_gather(const int* sIdx, int set,
                                                      const float* prims,
                                                      const float* ldsDst) {
  // Build the Tensor DMA descriptor (D#) in SGPRs and issue the gather of 16
  // codebook rows (16-bit row indices, 1024 x f32 each) into LDS.
  unsigned pk[8];
#pragma unroll
  for (int q = 0; q < 8; ++q) {
    const unsigned lo = (unsigned)sIdx[set * 16 + 2 * q];
    const unsigned hi = (unsigned)sIdx[set * 16 + 2 * q + 1];
    pk[q] = (unsigned)__builtin_amdgcn_readfirstlane((int)(lo | (hi << 16)));
  }
  const unsigned ldsA = (unsigned)__builtin_amdgcn_readfirstlane(
      (int)(unsigned)(uintptr_t)ldsDst);
  const unsigned long long ga = (unsigned long long)(uintptr_t)prims;
  const unsigned gaLo = (unsigned)__builtin_amdgcn_readfirstlane((int)(unsigned)ga);
  const unsigned gaHi = (unsigned)__builtin_amdgcn_readfirstlane((int)(unsigned)(ga >> 32));

  // D# group 0: count=1, gather_mode=1, 16-bit indices, lds_addr, global_addr, type=2
  v4u g0 = { 0x80000001u, ldsA, gaLo, (gaHi & 0x01FFFFFFu) | 0x80000000u };
  // D# group 1: data_size=4B, tensor_dim0=1024, tensor_dim1=1024, tile_dim0=1024,
  //             tile_dim1=16 (#indices), tensor_dim0_stride=1024
  v8i g1 = { (int)0x00020000u, (int)(1024u << 16), (int)(1024u << 16),
             (int)(1024u << 16), 16, 1024, 0, 0 };
  // D# groups 2+3: sixteen 16-bit row indices
  v4i g2 = { (int)pk[0], (int)pk[1], (int)pk[2], (int)pk[3] };
  v4i g3 = { (int)pk[4], (int)pk[5], (int)pk[6], (int)pk[7] };
  // trailing group: unused on this op (VADDR4 is "set to NULL" in the ISA) -> zeros
  v8i gz = { 0, 0, 0, 0, 0, 0, 0, 0 };
  __builtin_amdgcn_tensor_load_to_lds(g0, g1, g2, g3, gz, 0);
}

__global__ __launch_bounds__(256)
void vybn_codebook_kernel(const float* __restrict__ logits,
                          const float* __restrict__ prims,
                          float* __restrict__ outE,
                          int* __restrict__ outIdx,
                          int* __restrict__ counts) {
  // Double-buffered gather targets (2 x 64 KB) so TDM overlaps the WMMA stage.
  __shared__ float G[2][16 * DIM];
  __shared__ int   sIdx[32];      // selected primitive index per row (16 low + 16 high)

  const int tid = threadIdx.x;
  const int R   = blockIdx.x * 16;            // first low row of this block

  // ---------------- phase 1: gumbel noise + per-row argmax ----------------
  {
    const int p = tid >> 4;                    // row-pair 0..15
    const int i = tid & 15;                    // 16 threads cooperate per row-pair
    const int rowLo = R + p;
    const int rowHi = rowLo + ROWS_HALF;
    const float* Llo = logits + (size_t)rowLo * NPRIM;
    const float* Lhi = logits + (size_t)rowHi * NPRIM;
    const unsigned eBase = (unsigned)rowLo * NPRIM;

    float bvLo = -3.4e38f, bvHi = -3.4e38f;
    int   bkLo = 0,        bkHi = 0;

    for (int j = 0; j < 16; ++j) {
      const int kb = j * 64 + i * 4;
      // logits are streamed exactly once -> non-temporal, keep L2 for the codebook
      const v4f a = __builtin_nontemporal_load((const v4f*)(Llo + kb));
      const v4f b = __builtin_nontemporal_load((const v4f*)(Lhi + kb));
#pragma unroll
      for (int q = 0; q < 4; ++q) {
        const unsigned e0 = eBase + (unsigned)(kb + q);
        unsigned o0, o1;
        threefry2x32_42(e0, e0 + N_ELEMS_HALF, o0, o1);   // one cipher -> two rows' noise
        const float zlo = a[q] + gumbel_from_bits(o0);
        const float zhi = b[q] + gumbel_from_bits(o1);
        const int k = kb + q;                              // ascending per thread
        if (zlo > bvLo) { bvLo = zlo; bkLo = k; }
        if (zhi > bvHi) { bvHi = zhi; bkHi = k; }
      }
    }

    float* sv = &G[0][0];          // [512] values (lo:0..255, hi:256..511)
    int*   sk = (int*)&G[0][0] + 512;  // [512] arg indices
    sv[tid]       = bvLo;  sv[256 + tid] = bvHi;
    sk[tid]       = bkLo;  sk[256 + tid] = bkHi;
    __syncthreads();

    if (tid < 32) {
      const int set  = tid >> 4;   // 0 = low rows, 1 = high rows
      const int pp   = tid & 15;
      const int base = set * 256 + pp * 16;
      float bv = sv[base]; int bk = sk[base];
      for (int e = 1; e < 16; ++e) {
        const float v = sv[base + e]; const int kk = sk[base + e];
        if (v > bv || (v == bv && kk < bk)) { bv = v; bk = kk; }
      }
      const int gr = R + pp + set * ROWS_HALF;
      outIdx[gr] = bk;
      atomicAdd(&counts[bk], 1);
      sIdx[set * 16 + pp] = bk;
    }
    __syncthreads();
  }

  // ---------------- phases 2+3: double-buffered TDM gather + one-hot WMMA ----------------
  const int lane = tid & 31;
  const int wv   = tid >> 5;       // 8 waves
  const int m    = lane & 15;      // A: M row index; D: N column index
  const int h    = lane >> 4;      // half-wave selects K pair / M+8 block

  // A = identity one-hot over the 16 gathered rows (compacted one-hot GEMM), 4 K-steps of 4
  v2f afrag[4];
#pragma unroll
  for (int s = 0; s < 4; ++s) {
    const int k0 = s * 4 + 2 * h;
    afrag[s].x = (m == k0)     ? 1.0f : 0.0f;
    afrag[s].y = (m == k0 + 1) ? 1.0f : 0.0f;
  }

  // Wave 0 launches BOTH tile gathers up front; TDM ops from one wave complete
  // in order, so s_wait_tensorcnt(1) releases buffer 0 while buffer 1 streams.
  if (tid < 32) {
    vybn_issue_tdm_gather(sIdx, 0, prims, &G[0][0]);
    vybn_issue_tdm_gather(sIdx, 1, prims, &G[1][0]);
    __builtin_amdgcn_s_wait_tensorcnt(1);   // buffer 0 resident
  }
  __syncthreads();

  for (int set = 0; set < 2; ++set) {
    const float* Gs = &G[set][0];
    const int grBase = R + set * ROWS_HALF;
    // one-hot GEMM on the matrix pipe: D(16x16) = A(16x16 one-hot) x G-tile, exact in f32
    for (int t = 0; t < 8; ++t) {
      const int nt  = wv * 8 + t;       // 64 column tiles of 16
      const int col = nt * 16 + m;
      v8f c = {0.f, 0.f, 0.f, 0.f, 0.f, 0.f, 0.f, 0.f};
#pragma unroll
      for (int s = 0; s < 4; ++s) {
        const int k0 = s * 4 + 2 * h;
        v2f b;
        b.x = Gs[k0 * DIM + col];
        b.y = Gs[(k0 + 1) * DIM + col];
        c = __builtin_amdgcn_wmma_f32_16x16x4_f32(false, afrag[s], false, b,
                                                  (short)0, c, false, false);
      }
#pragma unroll
      for (int v = 0; v < 8; ++v) {
        // embeddings are written exactly once -> non-temporal store
        __builtin_nontemporal_store(c[v], &outE[(size_t)(grBase + v + 8 * h) * DIM + col]);
      }
    }
    if (set == 0) {
      if (tid < 32) __builtin_amdgcn_s_wait_tensorcnt(0);   // buffer 1 resident
      __syncthreads();
    }
  }
}

extern "C" void kernel_launch(void* const* d_in, const int* in_sizes, int n_in,
                              void* d_out, int out_size, void* d_ws, size_t ws_size,
                              hipStream_t stream) {
  const float* logits = (const float*)d_in[0];   // [8,4096,1024] f32
  const float* prims  = (const float*)d_in[1];   // [1024,1024] f32

  float* outE   = (float*)d_out;                                   // 33,554,432 f32
  int*   outIdx = (int*)(outE + (size_t)ROWS_TOTAL * DIM);         // 32,768 i32
  int*   counts = outIdx + ROWS_TOTAL;                             // 1,024 i32

  vybn_zero_counts<<<1, NPRIM, 0, stream>>>(counts);
  vybn_codebook_kernel<<<ROWS_HALF / 16, 256, 0, stream>>>(logits, prims, outE, outIdx, counts);
}